// HealEncoding_7017976562276
// MI455X (gfx1250) — compile-verified
//
#include <hip/hip_runtime.h>
#include <hip/hip_bf16.h>
#include <cstdint>

#define NLVL   8
#define NNEI   8
#define FDIM   4
#define NPTS   (1 << 20)
#define TBLN   196632        // 12 * ((2^7)^2 + 2)
#define THREADS 256

// ---------------------------------------------------------------------------
// CDNA5 async global->LDS copy (gfx1250) with graceful fallbacks.
//   mode 2: clang builtin       mode 1: inline asm       mode 0: plain copy
// The builtin's prototype (from clang's diagnostic) takes pointers to 16-byte
// int vectors: (addrspace(1) v4i*, addrspace(3) v4i*, imm offset, imm cpol).
// ---------------------------------------------------------------------------
#if defined(__gfx1250__)
  #if __has_builtin(__builtin_amdgcn_global_load_async_to_lds_b128)
    #define ASYNC_MODE 2
  #else
    #define ASYNC_MODE 1
  #endif
#else
  #define ASYNC_MODE 0
#endif

typedef int v4i_vec __attribute__((vector_size(16)));

__device__ __forceinline__ void async_copy_b128(void* lds_dst, const void* gsrc) {
#if ASYNC_MODE == 2
  __builtin_amdgcn_global_load_async_to_lds_b128(
      (__attribute__((address_space(1))) v4i_vec*)gsrc,
      (__attribute__((address_space(3))) v4i_vec*)lds_dst,
      /*imm offset*/ 0, /*cpol*/ 0);
#elif ASYNC_MODE == 1
  unsigned lds_off = (unsigned)(uintptr_t)(__attribute__((address_space(3))) void*)lds_dst;
  unsigned long long ga = (unsigned long long)gsrc;
  asm volatile("global_load_async_to_lds_b128 %0, %1, off"
               :: "v"(lds_off), "v"(ga) : "memory");
#else
  *(int4*)lds_dst = *(const int4*)gsrc;
#endif
}

template <int N>
__device__ __forceinline__ void wait_asynccnt() {
#if ASYNC_MODE == 2
  #if __has_builtin(__builtin_amdgcn_s_wait_asynccnt)
  __builtin_amdgcn_s_wait_asynccnt(N);
  #else
  asm volatile("s_wait_asynccnt %0" :: "i"(N) : "memory");
  #endif
#elif ASYNC_MODE == 1
  asm volatile("s_wait_asynccnt %0" :: "i"(N) : "memory");
#endif
}

// ---------------------------------------------------------------------------
// Hardware transcendentals.  v_sin_f32/v_cos_f32 take radians * 1/(2*pi)
// (1/(2*pi) is an inline constant on CDNA5, so each sine is 1 mul + 1 trans).
// ---------------------------------------------------------------------------
#define INV_2PI    0.15915494309189535f
#define HALF_2PI   0.07957747154594767f   // 0.5 * INV_2PI, folds the /2 in sin(d/2)
#define PI_OVER_2  1.5707963267948966f

__device__ __forceinline__ float hw_sin_half(float x_rad) {   // sin(x/2)
#if defined(__AMDGCN__)
  return __builtin_amdgcn_sinf(x_rad * HALF_2PI);
#else
  return __sinf(x_rad * 0.5f);
#endif
}
__device__ __forceinline__ float hw_cos(float x_rad) {
#if defined(__AMDGCN__)
  return __builtin_amdgcn_cosf(x_rad * INV_2PI);
#else
  return __cosf(x_rad);
#endif
}
__device__ __forceinline__ float hw_sqrt(float x) {
#if defined(__AMDGCN__)
  return __builtin_amdgcn_sqrtf(x);   // NaN for x<0, matching sqrt semantics
#else
  return sqrtf(x);
#endif
}
__device__ __forceinline__ float hw_rcp(float x) {
#if defined(__AMDGCN__)
  return __builtin_amdgcn_rcpf(x);
#else
  return 1.0f / x;
#endif
}

// atan2(y, x) for y,x >= 0 (y^2 + x^2 == 1 here).  NaN-propagating: the
// reference formula yields a<0 (-> NaN) when cos(colat1)*cos(colat2)<0, and
// jnp propagates that NaN; cndmask-based selection keeps NaN alive where
// fmin/fmax would drop it.  Degree-11 odd minimax on [0,1], ~1e-6 abs err.
__device__ __forceinline__ float atan2_nonneg(float y, float x) {
  bool  yg  = (y > x);                // false on NaN -> still computes with NaN
  float num = yg ? x : y;
  float den = yg ? y : x;             // den >= sqrt(0.5) > 0 when finite
  float r   = num * hw_rcp(den);
  float r2  = r * r;
  float p   = fmaf(r2, -0.01172120f, 0.05265332f);
  p = fmaf(r2, p, -0.11643287f);
  p = fmaf(r2, p,  0.19354346f);
  p = fmaf(r2, p, -0.33262347f);
  p = fmaf(r2, p,  0.99997726f);
  float at = r * p;
  return yg ? (PI_OVER_2 - at) : at;
}

// ---------------------------------------------------------------------------
// One thread per point; inner loop over the 8 levels.  Per level, this
// thread's 32B of neighbor indices + 64B of neighbor coords are streamed
// into LDS with the CDNA5 async engine, double-buffered across levels.
// Embedding gathers (25 MB table set, L2-resident) stay as direct b128 loads.
// ---------------------------------------------------------------------------
__global__ void __launch_bounds__(THREADS)
heal_encoding_kernel(const float* __restrict__ x,
                     const float* __restrict__ tables,
                     const int*   __restrict__ nbr_idx,
                     const int*   __restrict__ my_idx,
                     const float* __restrict__ nbr_crd,
                     float*       __restrict__ out) {
  __shared__ __align__(16) int   s_idx[2][THREADS][NNEI];        // 16 KB
  __shared__ __align__(16) float s_crd[2][THREADS][NNEI * 2];    // 32 KB

  const int tid = threadIdx.x;
  const int gid = blockIdx.x * THREADS + tid;

  // Point position: theta = pi/2 - deg2rad(lat), phi = deg2rad(lon)
  const float2 xv  = *(const float2*)(x + (size_t)gid * 2);
  const float DEG2RAD = 0.017453292519943295f;
  const float lat1 = PI_OVER_2 - xv.x * DEG2RAD;
  const float lon1 = xv.y * DEG2RAD;
  const float cos1 = hw_cos(lat1);

  auto prefetch = [&](int l, int b) {
    const int*   gi = nbr_idx + ((size_t)l * NPTS + (size_t)gid) * NNEI;       // 32B
    const float* gc = nbr_crd + ((size_t)l * NPTS + (size_t)gid) * NNEI * 2;   // 64B
    async_copy_b128(&s_idx[b][tid][0],  gi);
    async_copy_b128(&s_idx[b][tid][4],  gi + 4);
    async_copy_b128(&s_crd[b][tid][0],  gc);
    async_copy_b128(&s_crd[b][tid][4],  gc + 4);
    async_copy_b128(&s_crd[b][tid][8],  gc + 8);
    async_copy_b128(&s_crd[b][tid][12], gc + 12);
  };

  prefetch(0, 0);

  for (int l = 0; l < NLVL; ++l) {
    const int b = l & 1;
    if (l + 1 < NLVL) {
      prefetch(l + 1, (l + 1) & 1);   // issue next level first, then drain current
      wait_asynccnt<6>();             // in-order completion: first 6 (this level) done
    } else {
      wait_asynccnt<0>();
    }

    // Pull this thread's staged data out of LDS as b128s.
    const int4   i0 = *(const int4*)  &s_idx[b][tid][0];
    const int4   i1 = *(const int4*)  &s_idx[b][tid][4];
    const float4 c0 = *(const float4*)&s_crd[b][tid][0];
    const float4 c1 = *(const float4*)&s_crd[b][tid][4];
    const float4 c2 = *(const float4*)&s_crd[b][tid][8];
    const float4 c3 = *(const float4*)&s_crd[b][tid][12];

    const int   idxs[NNEI] = {i0.x, i0.y, i0.z, i0.w, i1.x, i1.y, i1.z, i1.w};
    const float la2[NNEI]  = {c0.x, c0.z, c1.x, c1.z, c2.x, c2.z, c3.x, c3.z};
    const float lo2[NNEI]  = {c0.y, c0.w, c1.y, c1.w, c2.y, c2.w, c3.y, c3.w};

    const float* tl = tables + (size_t)l * (TBLN * FDIM);

    // Issue all 9 L2 gathers early to hide latency behind the trig below.
    const int mi = my_idx[(size_t)l * NPTS + gid];
    float4 acc = *(const float4*)(tl + (size_t)mi * FDIM);      // my_emb
    float4 ev[NNEI];
#pragma unroll
    for (int k = 0; k < NNEI; ++k) {
      const int si = idxs[k] >= 0 ? idxs[k] : 0;
      ev[k] = *(const float4*)(tl + (size_t)si * FDIM);
    }

    // Great-circle distances (hardware trans ops).
    float d[NNEI];
    float dsum = 0.0f;
#pragma unroll
    for (int k = 0; k < NNEI; ++k) {
      const float sdlat = hw_sin_half(la2[k] - lat1);
      const float sdlon = hw_sin_half(lo2[k] - lon1);
      const float a  = fmaf(sdlat, sdlat, cos1 * hw_cos(la2[k]) * (sdlon * sdlon));
      const float t  = hw_sqrt(a);
      const float u  = hw_sqrt(1.0f - a);
      float dd = 2.0f * atan2_nonneg(t, u);
      dd = (idxs[k] >= 0) ? dd : 0.0f;    // where(valid, d, 0)
      d[k] = dd;
      dsum += dd;
    }

    const float rs = 1.0f / (dsum + 0.01f);
#pragma unroll
    for (int k = 0; k < NNEI; ++k) {
      const float w = d[k] * rs;
      acc.x = fmaf(w, ev[k].x, acc.x);
      acc.y = fmaf(w, ev[k].y, acc.y);
      acc.z = fmaf(w, ev[k].z, acc.z);
      acc.w = fmaf(w, ev[k].w, acc.w);
    }

    *(float4*)(out + (size_t)gid * (NLVL * FDIM) + l * FDIM) = acc;
  }
}

extern "C" void kernel_launch(void* const* d_in, const int* in_sizes, int n_in,
                              void* d_out, int out_size, void* d_ws, size_t ws_size,
                              hipStream_t stream) {
  // setup_inputs() order: x, tables, neighbor_index, my_index, neighbor_coords
  const float* x       = (const float*)d_in[0];
  const float* tables  = (const float*)d_in[1];
  const int*   nbr_idx = (const int*)  d_in[2];
  const int*   my_idx  = (const int*)  d_in[3];
  const float* nbr_crd = (const float*)d_in[4];
  float*       out     = (float*)d_out;

  const dim3 grid(NPTS / THREADS);
  const dim3 block(THREADS);
  hipLaunchKernelGGL(heal_encoding_kernel, grid, block, 0, stream,
                     x, tables, nbr_idx, my_idx, nbr_crd, out);
}